// Group_4303557230948
// MI455X (gfx1250) — compile-verified
//
#include <hip/hip_runtime.h>
#include <hip/hip_bf16.h>

// Problem constants (from reference setup_inputs): B=16, N=16384, G=256, K=32
#define N_PTS   16384
#define NGROUP  256
#define KNN     32

#define FPS_THREADS 1024
#define PTS_PER_THREAD (N_PTS / FPS_THREADS)   // 16

#define KNN_THREADS 256
#define KNN_WAVES   (KNN_THREADS / 32)         // 8
#define CT          16                          // centers per workgroup (one WMMA tile)
#define CHUNK       (N_PTS / KNN_WAVES)         // 2048 points per wave

typedef float v2f __attribute__((ext_vector_type(2)));
typedef float v8f __attribute__((ext_vector_type(8)));

__device__ __forceinline__ unsigned long long shfl_xor_u64(unsigned long long v, int m) {
    unsigned lo = (unsigned)(v & 0xFFFFFFFFull);
    unsigned hi = (unsigned)(v >> 32);
    lo = __shfl_xor(lo, m, 32);
    hi = __shfl_xor(hi, m, 32);
    return ((unsigned long long)hi << 32) | (unsigned long long)lo;
}

// ---------------------------------------------------------------------------
// Kernel 1: Farthest Point Sampling. One workgroup per batch.
// dists[N] lives entirely in LDS (64KB; CDNA5 WGP has 320KB).
// Emits centers (gathered from data_3d) straight into d_out's center section.
// argmax tie-break = lowest index (matches jnp.argmax) via key packing.
// ---------------------------------------------------------------------------
__global__ void fps_kernel(const float* __restrict__ xyz,
                           const float* __restrict__ data3d,
                           float* __restrict__ center_out) {
    extern __shared__ char fps_smem[];
    float* dists = (float*)fps_smem;                                   // N_PTS floats
    unsigned long long* wred = (unsigned long long*)(fps_smem + N_PTS * sizeof(float)); // 32 entries

    const int b = blockIdx.x;
    const float* P  = xyz    + (size_t)b * N_PTS * 3;
    const float* D3 = data3d + (size_t)b * N_PTS * 3;

    const int t = threadIdx.x;
    const int lane = t & 31;
    const int wave = t >> 5;   // 0..31

    // init dists = +inf (strided ownership: p = t + j*1024, LDS-bank friendly)
    for (int j = 0; j < PTS_PER_THREAD; ++j)
        dists[t + j * FPS_THREADS] = __int_as_float(0x7f800000);
    __syncthreads();

    int cur = 0;  // reference starts at index 0 and emits it as fidx[0]
    for (int g = 0; g < NGROUP; ++g) {
        if (t == 0) {
            size_t o = ((size_t)b * NGROUP + g) * 3;
            center_out[o + 0] = D3[(size_t)cur * 3 + 0];
            center_out[o + 1] = D3[(size_t)cur * 3 + 1];
            center_out[o + 2] = D3[(size_t)cur * 3 + 2];
        }
        const float cx = P[(size_t)cur * 3 + 0];
        const float cy = P[(size_t)cur * 3 + 1];
        const float cz = P[(size_t)cur * 3 + 2];

        float lmax = -1.0f;
        int   larg = 0;
        for (int j = 0; j < PTS_PER_THREAD; ++j) {
            int p = t + j * FPS_THREADS;              // ascending in p
            float dx = P[(size_t)p * 3 + 0] - cx;
            float dy = P[(size_t)p * 3 + 1] - cy;
            float dz = P[(size_t)p * 3 + 2] - cz;
            float d  = dx * dx + dy * dy + dz * dz;
            float dm = fminf(dists[p], d);
            dists[p] = dm;
            if (dm > lmax) { lmax = dm; larg = p; }   // strict > keeps first index
        }
        // key: max dist wins; on ties, smaller index wins (idx stored inverted)
        unsigned long long key =
            ((unsigned long long)__float_as_uint(lmax) << 32) |
            (unsigned long long)(0xFFFFFFFFu - (unsigned)larg);

        for (int off = 16; off > 0; off >>= 1) {
            unsigned long long o = shfl_xor_u64(key, off);
            if (o > key) key = o;
        }
        if (lane == 0) wred[wave] = key;
        __syncthreads();
        if (wave == 0) {
            unsigned long long k2 = wred[lane];       // 32 waves -> 32 lanes
            for (int off = 16; off > 0; off >>= 1) {
                unsigned long long o = shfl_xor_u64(k2, off);
                if (o > k2) k2 = o;
            }
            if (lane == 0) wred[0] = k2;
        }
        __syncthreads();
        unsigned long long win = wred[0];
        cur = (int)(0xFFFFFFFFu - (unsigned)(win & 0xFFFFFFFFull));
        __syncthreads();  // protect wred before next iteration's writes
    }
}

// ---------------------------------------------------------------------------
// Kernel 2: KNN + gather via V_WMMA_F32_16X16X4_F32.
// One workgroup = (batch, tile of 16 centers); 8 waves split N.
// d2 computed directly by WMMA: A-row=[x,y,z,|p|^2], B-col=[-2cx,-2cy,-2cz,1],
// C preloaded with |c|^2  =>  D[p][c] = |p-c|^2.
// D layout: lane holds one center column and 8 point rows -> per-lane streaming
// top-32 kept SORTED ASCENDING in 32 u64 REGISTERS; insertion is a fully
// unrolled, constant-indexed compare/select sweep (no dynamic indexing, so it
// never touches scratch), executed only under the rare (key < worst) mask.
// Then a 16-way shuffle-argmin merge per center.
// ---------------------------------------------------------------------------
__global__ void knn_kernel(const float* __restrict__ data3d,
                           const float* __restrict__ center_in,
                           float* __restrict__ neigh_out) {
    extern __shared__ char knn_smem[];
    unsigned long long* cand = (unsigned long long*)knn_smem;          // [CT][16][KNN] = 64KB
    int* sel = (int*)(knn_smem + (size_t)CT * 16 * KNN * sizeof(unsigned long long)); // [CT][KNN]

    const int b     = blockIdx.y;
    const int gbase = blockIdx.x * CT;
    const float* D3  = data3d   + (size_t)b * N_PTS * 3;
    const float* CEN = center_in + ((size_t)b * NGROUP + gbase) * 3;

    const int t    = threadIdx.x;
    const int lane = t & 31;
    const int wave = t >> 5;        // 0..7
    const int cidx = lane & 15;     // my center column within the tile
    const int hi   = lane >> 4;     // 0: K rows {0,1} / point rows 0-7; 1: K rows {2,3} / rows 8-15

    const float ccx = CEN[cidx * 3 + 0];
    const float ccy = CEN[cidx * 3 + 1];
    const float ccz = CEN[cidx * 3 + 2];
    const float cnorm = ccx * ccx + ccy * ccy + ccz * ccz;

    // B matrix 4x16 (VGPR0: K=0 lo-lanes / K=2 hi-lanes; VGPR1: K=1 / K=3)
    v2f Bv;
    Bv.x = hi ? (-2.0f * ccz) : (-2.0f * ccx);
    Bv.y = hi ? 1.0f          : (-2.0f * ccy);
    // C accumulator: |c|^2 broadcast over my 8 rows
    v8f Cv;
    #pragma unroll
    for (int k = 0; k < 8; ++k) Cv[k] = cnorm;

    // per-lane running top-32: sorted ascending u64 keys, ALL in registers
    unsigned long long top[KNN];
    #pragma unroll
    for (int j = 0; j < KNN; ++j) top[j] = ~0ull;
    unsigned long long worst = ~0ull;

    const int pbase0 = wave * CHUNK;
    for (int tb = 0; tb < CHUNK; tb += 16) {
        const int p = pbase0 + tb + cidx;               // point for my A-lane
        const float x = D3[(size_t)p * 3 + 0];
        const float y = D3[(size_t)p * 3 + 1];
        const float z = D3[(size_t)p * 3 + 2];
        const float pn = x * x + y * y + z * z;
        v2f Av;
        Av.x = hi ? z  : x;
        Av.y = hi ? pn : y;

        v8f Dv = __builtin_amdgcn_wmma_f32_16x16x4_f32(
            /*neg_a=*/false, Av, /*neg_b=*/false, Bv,
            /*c_mod=*/(short)0, Cv, /*reuse_a=*/false, /*reuse_b=*/false);

        const int ib = pbase0 + tb + hi * 8;            // my 8 point rows
        #pragma unroll
        for (int k = 0; k < 8; ++k) {
            float d = fmaxf(Dv[k], 0.0f);               // guard FP-negative self-distance
            unsigned long long key =
                ((unsigned long long)__float_as_uint(d) << 32) |
                (unsigned long long)(unsigned)(ib + k);
            if (key < worst) {
                // Branchless sorted insert, in place, descending j.
                // new[j] = top[j-1]          if top[j-1] >= key   (shift)
                //        = key               if top[j-1] < key <= top[j]
                //        = top[j]            if top[j] < key      (untouched)
                #pragma unroll
                for (int j = KNN - 1; j >= 1; --j) {
                    unsigned long long a = top[j - 1];
                    top[j] = (a >= key) ? a : ((top[j] >= key) ? key : top[j]);
                }
                top[0] = (top[0] >= key) ? key : top[0];
                worst = top[KNN - 1];
            }
        }
    }

    // dump my sorted list: list id = wave*2 + hi (16 lists per center)
    {
        unsigned long long* dst = cand + ((size_t)cidx * 16 + (wave * 2 + hi)) * KNN;
        #pragma unroll
        for (int j = 0; j < KNN; ++j) dst[j] = top[j];
    }
    __syncthreads();

    // 16-way merge per center; wave w handles centers w and w+8.
    // Keys are unique per center (each point index lives in exactly one list).
    for (int cc = wave; cc < CT; cc += KNN_WAVES) {
        int pos = 0;
        unsigned long long mykey =
            (lane < 16) ? cand[((size_t)cc * 16 + lane) * KNN] : ~0ull;
        for (int k = 0; k < KNN; ++k) {
            unsigned long long mn = mykey;
            for (int off = 16; off > 0; off >>= 1) {
                unsigned long long o = shfl_xor_u64(mn, off);
                if (o < mn) mn = o;
            }
            if (lane == 0) sel[cc * KNN + k] = (int)(unsigned)(mn & 0xFFFFFFFFull);
            if (mykey == mn) {
                ++pos;
                mykey = (lane < 16 && pos < KNN)
                          ? cand[((size_t)cc * 16 + lane) * KNN + pos] : ~0ull;
            }
        }
    }
    __syncthreads();

    // gather + center the neighborhoods: 16*32 = 512 (center,k) pairs
    for (int q = t; q < CT * KNN; q += KNN_THREADS) {
        const int c = q >> 5, k = q & 31;
        const int idx = sel[c * KNN + k];
        const int g = gbase + c;
        const float px = D3[(size_t)idx * 3 + 0];
        const float py = D3[(size_t)idx * 3 + 1];
        const float pz = D3[(size_t)idx * 3 + 2];
        const float gx = CEN[c * 3 + 0];
        const float gy = CEN[c * 3 + 1];
        const float gz = CEN[c * 3 + 2];
        size_t o = (((size_t)b * NGROUP + g) * KNN + k) * 3;
        neigh_out[o + 0] = px - gx;
        neigh_out[o + 1] = py - gy;
        neigh_out[o + 2] = pz - gz;
    }
}

// ---------------------------------------------------------------------------
extern "C" void kernel_launch(void* const* d_in, const int* in_sizes, int n_in,
                              void* d_out, int out_size, void* d_ws, size_t ws_size,
                              hipStream_t stream) {
    (void)n_in; (void)out_size; (void)d_ws; (void)ws_size;
    const float* xyz    = (const float*)d_in[0];
    const float* data3d = (const float*)d_in[1];
    const int B = in_sizes[0] / (N_PTS * 3);   // 16

    float* out    = (float*)d_out;
    float* neigh  = out;                                        // [B,G,K,3]
    float* center = out + (size_t)B * NGROUP * KNN * 3;         // [B,G,3]

    const size_t fps_lds = (size_t)N_PTS * sizeof(float) + 32 * sizeof(unsigned long long) + 64;
    fps_kernel<<<dim3(B), dim3(FPS_THREADS), fps_lds, stream>>>(xyz, data3d, center);

    const size_t knn_lds = (size_t)CT * 16 * KNN * sizeof(unsigned long long)
                         + (size_t)CT * KNN * sizeof(int);
    knn_kernel<<<dim3(NGROUP / CT, B), dim3(KNN_THREADS), knn_lds, stream>>>(data3d, center, neigh);
}